// MultilayerGCN_13211319402817
// MI455X (gfx1250) — compile-verified
//
#include <hip/hip_runtime.h>

// ---------------------------------------------------------------------------
// 3-layer GCN for MI455X (gfx1250, wave32).
//  - degree norms via int atomics, CSR-by-dst built once (scan + scatter)
//  - aggregation: one wave32 per dst node, float4 register accumulation,
//    coalesced 1KB row gathers (activations are 102MB -> resident in 192MB L2),
//    global_prefetch_b8 one edge ahead
//  - GEMM: V_WMMA_F32_16X16X4_F32, 16xK A-strip in LDS (stride K+4, no bank
//    conflicts), W accessed via immediate-offset loads (NOUT is a template
//    constant; W2 zero-padded 40->48 cols so there is no load predication),
//    dual accumulators to break the WMMA RAW chain, bias+BN+ReLU fused.
// ---------------------------------------------------------------------------

typedef __attribute__((ext_vector_type(2))) float v2f;
typedef __attribute__((ext_vector_type(8))) float v8f;

#define BN_EPS 1e-5f

__global__ void zero_i32(int* __restrict__ p, int n) {
    int i = blockIdx.x * blockDim.x + threadIdx.x;
    if (i < n) p[i] = 0;
}

__global__ void deg_count(const int* __restrict__ src, const int* __restrict__ dst,
                          int* __restrict__ degs, int* __restrict__ degd, int e) {
    int i = blockIdx.x * blockDim.x + threadIdx.x;
    if (i < e) {
        atomicAdd(&degs[src[i]], 1);
        atomicAdd(&degd[dst[i]], 1);
    }
}

__global__ void deg_norm(const int* __restrict__ degs, const int* __restrict__ degd,
                         float* __restrict__ ns, float* __restrict__ nd, int n) {
    int i = blockIdx.x * blockDim.x + threadIdx.x;
    if (i < n) {
        int a = degs[i];
        int b = degd[i];
        ns[i] = (a > 0) ? rsqrtf((float)a) : 0.0f;
        nd[i] = (b > 0) ? rsqrtf((float)b) : 0.0f;
    }
}

// Exclusive scan of degd (length ndeg, virtual 0-pad) into out (length ntot=ndeg+1).
__global__ void scan_block(const int* __restrict__ deg, int* __restrict__ out,
                           int* __restrict__ bsum, int ndeg, int ntot) {
    __shared__ int sh[256];
    int i = blockIdx.x * 256 + threadIdx.x;
    int v = (i < ndeg) ? deg[i] : 0;
    sh[threadIdx.x] = v;
    __syncthreads();
    for (int off = 1; off < 256; off <<= 1) {
        int t = (threadIdx.x >= (unsigned)off) ? sh[threadIdx.x - off] : 0;
        __syncthreads();
        sh[threadIdx.x] += t;
        __syncthreads();
    }
    if (i < ntot) out[i] = sh[threadIdx.x] - v;   // exclusive
    if (threadIdx.x == 255) bsum[blockIdx.x] = sh[255];
}

__global__ void scan_tops(int* __restrict__ bsum, int nb) {
    if (blockIdx.x == 0 && threadIdx.x == 0) {
        int acc = 0;
        for (int i = 0; i < nb; ++i) { int t = bsum[i]; bsum[i] = acc; acc += t; }
    }
}

__global__ void scan_fix(int* __restrict__ out, int* __restrict__ cursor,
                         const int* __restrict__ bsum, int n, int ntot) {
    int i = blockIdx.x * blockDim.x + threadIdx.x;
    if (i < ntot) {
        int v = out[i] + bsum[i >> 8];
        out[i] = v;
        if (i < n) cursor[i] = v;
    }
}

__global__ void csr_fill(const int* __restrict__ src, const int* __restrict__ dst,
                         int* __restrict__ cursor, int* __restrict__ col, int e) {
    int i = blockIdx.x * blockDim.x + threadIdx.x;
    if (i < e) {
        int p = atomicAdd(&cursor[dst[i]], 1);
        col[p] = src[i];
    }
}

// Zero-pad W [K x Nin] -> Wp [K x Npad], bias -> bp (removes all load
// predication from the Fout=40 GEMM).
__global__ void pad_w(const float* __restrict__ W, const float* __restrict__ b,
                      float* __restrict__ Wp, float* __restrict__ bp,
                      int K, int Nin, int Npad) {
    int i = blockIdx.x * blockDim.x + threadIdx.x;
    if (i < K * Npad) {
        int k = i / Npad, n = i % Npad;
        Wp[i] = (n < Nin) ? W[k * Nin + n] : 0.0f;
    }
    if (i < Npad) bp[i] = (i < Nin) ? b[i] : 0.0f;
}

// out[v,:] = norm_dst[v] * sum_{u in in(v)} norm_src[u] * h[u,:]
// One wave32 per destination node; float4 (b128) gathers, F/128 vec-accums.
template <int F>
__global__ __launch_bounds__(256)
void gcn_aggregate(const float* __restrict__ h, const int* __restrict__ rowptr,
                   const int* __restrict__ col, const float* __restrict__ nsrc,
                   const float* __restrict__ ndst, float* __restrict__ out, int n) {
    constexpr int V = F / 128;     // float4s per lane (1 or 2)
    int wave = (blockIdx.x * blockDim.x + threadIdx.x) >> 5;
    int lane = threadIdx.x & 31;
    if (wave >= n) return;

    float4 acc[V];
#pragma unroll
    for (int k = 0; k < V; ++k) acc[k] = make_float4(0.f, 0.f, 0.f, 0.f);

    int beg = rowptr[wave], end = rowptr[wave + 1];
    for (int j = beg; j < end; ++j) {
        int s = col[j];
        if (j + 1 < end)  // global_prefetch_b8: hide gather latency
            __builtin_prefetch(h + (size_t)col[j + 1] * F + lane * 4, 0, 0);
        float w = nsrc[s];
        const float4* row = (const float4*)(h + (size_t)s * F);
#pragma unroll
        for (int k = 0; k < V; ++k) {
            float4 r = row[lane + 32 * k];
            acc[k].x = fmaf(w, r.x, acc[k].x);
            acc[k].y = fmaf(w, r.y, acc[k].y);
            acc[k].z = fmaf(w, r.z, acc[k].z);
            acc[k].w = fmaf(w, r.w, acc[k].w);
        }
    }

    float wd = ndst[wave];
    float4* o = (float4*)(out + (size_t)wave * F);
#pragma unroll
    for (int k = 0; k < V; ++k)
        o[lane + 32 * k] = make_float4(acc[k].x * wd, acc[k].y * wd,
                                       acc[k].z * wd, acc[k].w * wd);
}

// C[16x16 tile] = A[16xK] @ W[KxNOUT] + bias, optional BN+ReLU.
// 128 threads = 4 waves per block; A strip shared via LDS (padded stride).
// NOUT compile-time -> all W loads are base+imm-offset, zero loop address ALU.
template <int K, int NOUT, bool BN>
__global__ __launch_bounds__(128)
void gemm_wmma(const float* __restrict__ A, const float* __restrict__ W,
               const float* __restrict__ bias,
               const float* __restrict__ g, const float* __restrict__ bt,
               const float* __restrict__ mu, const float* __restrict__ vr,
               float* __restrict__ out, int nst) {
    constexpr int LDA = K + 4;             // dodge 64-bank conflicts (K%64==0)
    __shared__ float As[16 * LDA];

    int mtile = blockIdx.x;
    int wv    = threadIdx.x >> 5;
    int lane  = threadIdx.x & 31;

    // Cooperative, coalesced float4 staging of the 16xK A strip.
    const float* Ab = A + (size_t)mtile * (16 * K);
    for (int i = threadIdx.x; i < 16 * K / 4; i += 128) {
        int r = i / (K / 4);
        int c = i % (K / 4);
        float4 v = ((const float4*)Ab)[i];
        *(float4*)&As[r * LDA + c * 4] = v;
    }
    __syncthreads();

    int ntile = blockIdx.y * 4 + wv;
    int ncol0 = ntile * 16;
    if (ncol0 >= NOUT) return;             // wave-uniform; after the only sync

    int m  = lane & 15;                    // A row / B column within tile
    int kh = (lane >> 4) << 1;             // K sub-offset: 0 (lanes 0-15) or 2
    int n  = ncol0 + m;

    const float* Wp  = W + (kh * NOUT + n);   // lane base; loop uses imm offsets
    const float* Ap  = &As[m * LDA + kh];

    v8f c0 = {0.f, 0.f, 0.f, 0.f, 0.f, 0.f, 0.f, 0.f};
    v8f c1 = {0.f, 0.f, 0.f, 0.f, 0.f, 0.f, 0.f, 0.f};
#pragma unroll
    for (int kb = 0; kb < K; kb += 8) {
        v2f a0, b0, a1, b1;
        a0.x = Ap[kb];                      // ds_load_b64 (8B aligned)
        a0.y = Ap[kb + 1];
        b0.x = Wp[(kb + 0) * NOUT];         // global_load_b32 base+imm
        b0.y = Wp[(kb + 1) * NOUT];
        c0 = __builtin_amdgcn_wmma_f32_16x16x4_f32(
                 false, a0, false, b0, (short)0, c0, false, false);
        a1.x = Ap[kb + 4];
        a1.y = Ap[kb + 5];
        b1.x = Wp[(kb + 4) * NOUT];
        b1.y = Wp[(kb + 5) * NOUT];
        c1 = __builtin_amdgcn_wmma_f32_16x16x4_f32(
                 false, a1, false, b1, (short)0, c1, false, false);
    }

    float bi = bias[n];
    float gg = 1.f, bb = 0.f, mm = 0.f, iv = 1.f;
    if (BN) { gg = g[n]; bb = bt[n]; mm = mu[n]; iv = rsqrtf(vr[n] + BN_EPS); }

    if (n >= nst) return;                  // only the padded tail lanes exit
    int rbase = mtile * 16 + ((lane >> 4) << 3);   // rows 0-7 or 8-15 of tile
    float* op = out + (size_t)rbase * nst + n;
#pragma unroll
    for (int i = 0; i < 8; ++i) {
        float v = (c0[i] + c1[i]) + bi;
        if (BN) { v = (v - mm) * iv * gg + bb; v = fmaxf(v, 0.0f); }
        op[i * nst] = v;
    }
}

extern "C" void kernel_launch(void* const* d_in, const int* in_sizes, int n_in,
                              void* d_out, int out_size, void* d_ws, size_t ws_size,
                              hipStream_t stream) {
    const float* x  = (const float*)d_in[0];
    const int*   src = (const int*)d_in[1];
    const int*   dst = (const int*)d_in[2];
    const float* W0 = (const float*)d_in[3];  const float* b0 = (const float*)d_in[4];
    const float* W1 = (const float*)d_in[5];  const float* b1 = (const float*)d_in[6];
    const float* W2 = (const float*)d_in[7];  const float* b2 = (const float*)d_in[8];
    const float* g0 = (const float*)d_in[9];  const float* be0 = (const float*)d_in[10];
    const float* m0 = (const float*)d_in[11]; const float* v0 = (const float*)d_in[12];
    const float* g1 = (const float*)d_in[13]; const float* be1 = (const float*)d_in[14];
    const float* m1 = (const float*)d_in[15]; const float* v1 = (const float*)d_in[16];

    const int E  = in_sizes[1];
    const int H  = in_sizes[4];          // 256
    const int F2 = in_sizes[8];          // 40
    const int K0 = in_sizes[3] / H;      // 128
    const int N  = in_sizes[0] / K0;     // 100000 (divisible by 16)
    const int F2P = 48;                  // padded output width for layer 2

    // Workspace carve-out (256B aligned). Total ~220MB.
    char* ws = (char*)d_ws;
    size_t off = 0;
    auto take = [&](size_t bytes) -> char* {
        char* p = ws + off;
        off = (off + bytes + 255) & ~(size_t)255;
        return p;
    };
    int*   degs   = (int*)take((size_t)N * 4);
    int*   degd   = (int*)take((size_t)N * 4);
    int*   rowptr = (int*)take((size_t)(N + 1) * 4);
    int*   cursor = (int*)take((size_t)N * 4);
    int    nsb    = (N + 1 + 255) / 256;
    int*   bsum   = (int*)take((size_t)nsb * 4);
    float* ns     = (float*)take((size_t)N * 4);
    float* nd     = (float*)take((size_t)N * 4);
    float* W2p    = (float*)take((size_t)H * F2P * 4);
    float* b2p    = (float*)take((size_t)F2P * 4);
    int*   col    = (int*)take((size_t)E * 4);
    float* bufA   = (float*)take((size_t)N * H * 4);
    float* bufB   = (float*)take((size_t)N * H * 4);

    const int TB = 256;
    dim3 eb((E + TB - 1) / TB), nb((N + TB - 1) / TB);

    // --- graph preprocessing (degrees, norms, CSR by dst; reused 3x) ---
    zero_i32<<<nb, TB, 0, stream>>>(degs, N);
    zero_i32<<<nb, TB, 0, stream>>>(degd, N);
    deg_count<<<eb, TB, 0, stream>>>(src, dst, degs, degd, E);
    deg_norm<<<nb, TB, 0, stream>>>(degs, degd, ns, nd, N);
    scan_block<<<nsb, 256, 0, stream>>>(degd, rowptr, bsum, N, N + 1);
    scan_tops<<<1, 32, 0, stream>>>(bsum, nsb);
    scan_fix<<<nsb, 256, 0, stream>>>(rowptr, cursor, bsum, N, N + 1);
    csr_fill<<<eb, TB, 0, stream>>>(src, dst, cursor, col, E);
    pad_w<<<(H * F2P + TB - 1) / TB, TB, 0, stream>>>(W2, b2, W2p, b2p, H, F2, F2P);

    dim3 ab((N + 7) / 8);                        // 8 waves per 256-thread block
    int mtiles = N / 16;
    dim3 gH(mtiles, (256 / 16) / 4);             // 256 cols -> grid.y = 4
    dim3 g2(mtiles, 1);                          // 48 padded cols -> 3 tiles

    // --- layer 0: agg(x) -> bufA[128], gemm+BN+ReLU -> bufB[256] ---
    gcn_aggregate<128><<<ab, TB, 0, stream>>>(x, rowptr, col, ns, nd, bufA, N);
    gemm_wmma<128, 256, true><<<gH, 128, 0, stream>>>(bufA, W0, b0, g0, be0, m0, v0,
                                                      bufB, H);

    // --- layer 1 ---
    gcn_aggregate<256><<<ab, TB, 0, stream>>>(bufB, rowptr, col, ns, nd, bufA, N);
    gemm_wmma<256, 256, true><<<gH, 128, 0, stream>>>(bufA, W1, b1, g1, be1, m1, v1,
                                                      bufB, H);

    // --- layer 2 (no BN/ReLU, padded W) -> d_out [N x 40] ---
    gcn_aggregate<256><<<ab, TB, 0, stream>>>(bufB, rowptr, col, ns, nd, bufA, N);
    gemm_wmma<256, 48, false><<<g2, 128, 0, stream>>>(bufA, W2p, b2p, g0, be0, m0, v0,
                                                      (float*)d_out, F2);
}